// ContourPointGCN_4638564679870
// MI455X (gfx1250) — compile-verified
//
#include <hip/hip_runtime.h>
#include <hip/hip_bf16.h>

// Problem constants (match reference)
#define BB   4
#define CC   256
#define HWSZ 65536           // 256*256
#define PP   256
#define THRESH 0.8f
#define EPSV 1e-5f

typedef __attribute__((ext_vector_type(2))) float v2f;
typedef __attribute__((ext_vector_type(8))) float v8f;

// ---------------------------------------------------------------------------
// Kernel 1: bulk copy x -> out (dominant cost: 512 MB of HBM traffic)
// ---------------------------------------------------------------------------
__global__ __launch_bounds__(256) void cpg_copy_kernel(const float4* __restrict__ x,
                                                       float4* __restrict__ out, int n4) {
    int i = blockIdx.x * 256 + threadIdx.x;
    if (i < n4) out[i] = x[i];
}

// ---------------------------------------------------------------------------
// Kernel 2: per-batch top-256 via 3-level radix select on float bits
// (positive floats: bit pattern is monotonic). Produces indices sorted by
// (value desc, index asc) to match jax.lax.top_k ordering exactly.
// One workgroup (1024 threads) per batch image.
// ---------------------------------------------------------------------------
__device__ __forceinline__ unsigned th_bits(float v) {
    v = (v < THRESH) ? 0.0f : v;
    return __float_as_uint(v);
}

__global__ __launch_bounds__(1024) void cpg_topk_kernel(const float* __restrict__ edge,
                                                        int* __restrict__ idx_out) {
    const int b = blockIdx.x;
    const int t = threadIdx.x;
    const float* e = edge + (size_t)b * HWSZ;
    const int base = t * 64;                       // contiguous chunk per thread (index order)

    __shared__ unsigned hist[4096];
    __shared__ unsigned tcnt[1024];
    __shared__ unsigned long long cand[PP];
    __shared__ unsigned sh_sel[3];
    __shared__ int      sh_need[3];
    __shared__ unsigned candCount;

    // ---- level 1: bins = bits >> 20 (values in [0,1) -> bin <= 1015) ----
    for (int i = t; i < 4096; i += 1024) hist[i] = 0u;
    if (t == 0) candCount = 0u;
    __syncthreads();
    for (int i = 0; i < 64; ++i) atomicAdd(&hist[th_bits(e[base + i]) >> 20], 1u);
    __syncthreads();
    if (t == 0) {
        int need = PP; int bin = 1023;
        for (;;) { unsigned c = hist[bin]; if ((int)c >= need) break; need -= (int)c; --bin; }
        sh_sel[0] = (unsigned)bin; sh_need[0] = need;
    }
    __syncthreads();
    const unsigned sel1 = sh_sel[0]; const int need1 = sh_need[0];

    // ---- level 2: bits[19:8] within the pivot level-1 bin ----
    for (int i = t; i < 4096; i += 1024) hist[i] = 0u;
    __syncthreads();
    for (int i = 0; i < 64; ++i) {
        unsigned bits = th_bits(e[base + i]);
        if ((bits >> 20) == sel1) atomicAdd(&hist[(bits >> 8) & 0xFFFu], 1u);
    }
    __syncthreads();
    if (t == 0) {
        int need = need1; int bin = 4095;
        for (;;) { unsigned c = hist[bin]; if ((int)c >= need) break; need -= (int)c; --bin; }
        sh_sel[1] = (unsigned)bin; sh_need[1] = need;
    }
    __syncthreads();
    const unsigned sel2 = sh_sel[1]; const int need2 = sh_need[1];

    // ---- level 3: bits[7:0] -> exact pivot value ----
    for (int i = t; i < 4096; i += 1024) hist[i] = 0u;
    __syncthreads();
    for (int i = 0; i < 64; ++i) {
        unsigned bits = th_bits(e[base + i]);
        if ((bits >> 20) == sel1 && ((bits >> 8) & 0xFFFu) == sel2)
            atomicAdd(&hist[bits & 0xFFu], 1u);
    }
    __syncthreads();
    if (t == 0) {
        int need = need2; int bin = 255;
        for (;;) { unsigned c = hist[bin]; if ((int)c >= need) break; need -= (int)c; --bin; }
        sh_sel[2] = (unsigned)bin; sh_need[2] = need;
    }
    __syncthreads();
    const unsigned pivot = (sel1 << 20) | (sel2 << 8) | sh_sel[2];
    const int needTies = sh_need[2];               // # of pivot-valued elems to take, index order

    // ---- deterministic tie ranking in global index order ----
    unsigned lt = 0;
    for (int i = 0; i < 64; ++i) if (th_bits(e[base + i]) == pivot) ++lt;
    tcnt[t] = lt;
    __syncthreads();
    if (t == 0) { unsigned run = 0; for (int i = 0; i < 1024; ++i) { unsigned c = tcnt[i]; tcnt[i] = run; run += c; } }
    __syncthreads();
    unsigned run = tcnt[t];

    // ---- collect exactly 256 selected (value,index) keys ----
    for (int i = 0; i < 64; ++i) {
        unsigned bits = th_bits(e[base + i]);
        bool sel = false;
        if (bits > pivot) sel = true;
        else if (bits == pivot) { if ((int)run < needTies) sel = true; ++run; }
        if (sel) {
            unsigned pos = atomicAdd(&candCount, 1u);
            if (pos < PP)
                cand[pos] = ((unsigned long long)bits << 32) |
                            (unsigned long long)(0xFFFFFFFFu - (unsigned)(base + i));
        }
    }
    __syncthreads();

    // ---- bitonic sort 256 keys descending: (value desc, index asc) ----
    for (int ksz = 2; ksz <= PP; ksz <<= 1) {
        for (int j = ksz >> 1; j > 0; j >>= 1) {
            if (t < PP) {
                int ixj = t ^ j;
                if (ixj > t) {
                    unsigned long long a = cand[t], c2 = cand[ixj];
                    bool descBlock = ((t & ksz) == 0);
                    bool doSwap = descBlock ? (a < c2) : (a > c2);
                    if (doSwap) { cand[t] = c2; cand[ixj] = a; }
                }
            }
            __syncthreads();
        }
    }
    if (t < PP)
        idx_out[b * PP + t] = (int)(0xFFFFFFFFu - (unsigned)(cand[t] & 0xFFFFFFFFull));
}

// ---------------------------------------------------------------------------
// Kernel 3: gather feat[b][p][c] = x[b][c][idx[b][p]]   (feat row-major P x C)
// ---------------------------------------------------------------------------
__global__ __launch_bounds__(256) void cpg_gather_kernel(const float* __restrict__ x,
                                                         const int* __restrict__ idx,
                                                         float* __restrict__ feat) {
    int bp = blockIdx.x;
    int b = bp >> 8;               // / PP
    int p = bp & 255;
    int c = threadIdx.x;
    int pos = idx[b * PP + p];
    feat[(((size_t)b * PP) + p) * CC + c] = x[(((size_t)b * CC) + c) * HWSZ + pos];
}

// ---------------------------------------------------------------------------
// Kernel 4: stage 1 GEMM  z1 = relu(BN(w_adj @ feat)) + feat   per batch.
// One 16x16 output tile per wave, V_WMMA_F32_16X16X4_F32, K=256 in steps of 4.
// ---------------------------------------------------------------------------
__global__ __launch_bounds__(256) void cpg_stage1_kernel(const float* __restrict__ w_adj,
                                                         const float* __restrict__ feat,
                                                         const float* __restrict__ g,
                                                         const float* __restrict__ bt,
                                                         const float* __restrict__ mn,
                                                         const float* __restrict__ vr,
                                                         float* __restrict__ z1) {
    const int wave = threadIdx.x >> 5;
    const int lane = threadIdx.x & 31;
    const int b    = blockIdx.x >> 5;      // 32 blocks per batch
    const int blk  = blockIdx.x & 31;
    const int tile = blk * 8 + wave;       // 0..255
    const int i0 = (tile >> 4) * 16;       // node-row block
    const int d0 = (tile & 15) * 16;       // feature-col block
    const int lhalf = lane >> 4;           // 0: K{0,1}, 1: K{2,3}
    const int lm = lane & 15;

    const float* featB = feat + (size_t)b * PP * CC;
    v8f acc = {};
    for (int k0 = 0; k0 < PP; k0 += 4) {
        int ka = k0 + lhalf * 2;
        v2f a, bm;
        a.x  = w_adj[(i0 + lm) * PP + ka];         // A 16x4: lane=M, halves hold K pairs
        a.y  = w_adj[(i0 + lm) * PP + ka + 1];
        bm.x = featB[ka * CC + d0 + lm];           // B 4x16: lane=N, halves split K
        bm.y = featB[(ka + 1) * CC + d0 + lm];
        acc = __builtin_amdgcn_wmma_f32_16x16x4_f32(false, a, false, bm,
                                                    (short)0, acc, false, false);
    }
    // epilogue: BN over node dim (rows) + ReLU + residual
    #pragma unroll
    for (int r = 0; r < 8; ++r) {
        int i = i0 + r + lhalf * 8;                // D layout: M = r + 8*(lane>=16)
        int d = d0 + lm;                           // N = lane & 15
        float inv  = g[i] * rsqrtf(vr[i] + EPSV);
        float val  = acc[r] * inv + (bt[i] - mn[i] * inv);
        val = fmaxf(val, 0.0f) + featB[(size_t)i * CC + d];
        z1[(((size_t)b * PP) + i) * CC + d] = val;
    }
}

// ---------------------------------------------------------------------------
// Kernel 5: stage 2 GEMM  z2 = relu(BN(w_wg @ z1^T)) fused with scatter into out.
// ---------------------------------------------------------------------------
__global__ __launch_bounds__(256) void cpg_stage2_kernel(const float* __restrict__ w_wg,
                                                         const float* __restrict__ z1,
                                                         const float* __restrict__ g,
                                                         const float* __restrict__ bt,
                                                         const float* __restrict__ mn,
                                                         const float* __restrict__ vr,
                                                         const int* __restrict__ idx,
                                                         float* __restrict__ out) {
    const int wave = threadIdx.x >> 5;
    const int lane = threadIdx.x & 31;
    const int b    = blockIdx.x >> 5;
    const int blk  = blockIdx.x & 31;
    const int tile = blk * 8 + wave;
    const int c0 = (tile >> 4) * 16;       // channel-row block
    const int p0 = (tile & 15) * 16;       // point-col block
    const int lhalf = lane >> 4;
    const int lm = lane & 15;

    const float* z1B = z1 + (size_t)b * PP * CC;
    v8f acc = {};
    for (int k0 = 0; k0 < CC; k0 += 4) {
        int ka = k0 + lhalf * 2;
        v2f a, bm;
        a.x  = w_wg[(c0 + lm) * CC + ka];
        a.y  = w_wg[(c0 + lm) * CC + ka + 1];
        bm.x = z1B[(p0 + lm) * CC + ka];           // B[k][n] = z1[p0+n][k]
        bm.y = z1B[(p0 + lm) * CC + ka + 1];
        acc = __builtin_amdgcn_wmma_f32_16x16x4_f32(false, a, false, bm,
                                                    (short)0, acc, false, false);
    }
    #pragma unroll
    for (int r = 0; r < 8; ++r) {
        int c = c0 + r + lhalf * 8;
        int p = p0 + lm;
        float inv = g[c] * rsqrtf(vr[c] + EPSV);
        float val = acc[r] * inv + (bt[c] - mn[c] * inv);
        val = fmaxf(val, 0.0f);
        int pos = idx[b * PP + p];
        out[(((size_t)b * CC) + c) * HWSZ + pos] = val;
    }
}

// ---------------------------------------------------------------------------
extern "C" void kernel_launch(void* const* d_in, const int* in_sizes, int n_in,
                              void* d_out, int out_size, void* d_ws, size_t ws_size,
                              hipStream_t stream) {
    const float* x     = (const float*)d_in[0];
    const float* edge  = (const float*)d_in[1];
    const float* w_adj = (const float*)d_in[2];
    const float* g_adj = (const float*)d_in[3];
    const float* b_adj = (const float*)d_in[4];
    const float* m_adj = (const float*)d_in[5];
    const float* v_adj = (const float*)d_in[6];
    const float* w_wg  = (const float*)d_in[7];
    const float* g_wg  = (const float*)d_in[8];
    const float* b_wg  = (const float*)d_in[9];
    const float* m_wg  = (const float*)d_in[10];
    const float* v_wg  = (const float*)d_in[11];
    float* out = (float*)d_out;

    // workspace layout: idx (4 KB) | feat (1 MB) | z1 (1 MB)
    int*   idx_ws  = (int*)d_ws;
    float* feat_ws = (float*)((char*)d_ws + 4096);
    float* z1_ws   = feat_ws + (size_t)BB * PP * CC;

    // 1) bulk copy x -> out (float4 wide, 16M vectors)
    const int n4 = (BB * CC * HWSZ) / 4;
    cpg_copy_kernel<<<n4 / 256, 256, 0, stream>>>((const float4*)x, (float4*)out, n4);

    // 2) per-batch top-256 (sorted, jax-compatible tie-break)
    cpg_topk_kernel<<<BB, 1024, 0, stream>>>(edge, idx_ws);

    // 3) gather point features (B,P,C)
    cpg_gather_kernel<<<BB * PP, CC, 0, stream>>>(x, idx_ws, feat_ws);

    // 4) stage-1 GCN: adjacency mix + BN + ReLU + residual (WMMA f32)
    cpg_stage1_kernel<<<BB * 32, 256, 0, stream>>>(w_adj, feat_ws,
                                                   g_adj, b_adj, m_adj, v_adj, z1_ws);

    // 5) stage-2 GCN: feature transform + BN + ReLU, fused scatter (WMMA f32)
    cpg_stage2_kernel<<<BB * 32, 256, 0, stream>>>(w_wg, z1_ws,
                                                   g_wg, b_wg, m_wg, v_wg, idx_ws, out);
}